// GraphConvolution_67723044323421
// MI455X (gfx1250) — compile-verified
//
#include <hip/hip_runtime.h>

// ---------------------------------------------------------------------------
// GCN layer for MI455X (gfx1250, wave32):
//   support = segment_sum(x[src] * w, dst)      (edge-parallel fp32 atomics)
//   out     = support @ weight + bias           (V_WMMA_F32_16X16X4_F32)
// ---------------------------------------------------------------------------

#define N_NODES 10000
#define D_IN    128
#define D_OUT   128

typedef float v2f __attribute__((ext_vector_type(2)));
typedef float v8f __attribute__((ext_vector_type(8)));

// ---------------------------------------------------------------------------
// Stage 1+2: one wave32 per edge; lane l handles features [4l, 4l+4).
// x and support both fit in L2 (5.12 MB each on a 192 MB L2), so this is
// L2-atomic-throughput bound; use hardware global_atomic_add_f32.
// ---------------------------------------------------------------------------
__global__ __launch_bounds__(256)
void gcn_scatter_kernel(const float* __restrict__ x,
                        const int*   __restrict__ src,
                        const int*   __restrict__ dst,
                        const float* __restrict__ w,
                        float*       __restrict__ support,
                        int n_edges) {
    int tid  = blockIdx.x * blockDim.x + threadIdx.x;
    int e    = tid >> 5;          // edge index (one wave per edge)
    int lane = tid & 31;
    if (e >= n_edges) return;

    int   s  = src[e];
    int   d  = dst[e];
    float wt = w[e];

    const float4* xv4 = reinterpret_cast<const float4*>(x + (size_t)s * D_IN);
    float4 v = xv4[lane];

    float* sp = support + (size_t)d * D_IN + lane * 4;
    // unsafeAtomicAdd -> global_atomic_add_f32 (no CAS loop)
    unsafeAtomicAdd(sp + 0, v.x * wt);
    unsafeAtomicAdd(sp + 1, v.y * wt);
    unsafeAtomicAdd(sp + 2, v.z * wt);
    unsafeAtomicAdd(sp + 3, v.w * wt);
}

// ---------------------------------------------------------------------------
// Stage 3: [10000 x 128] @ [128 x 128] + bias, full fp32 via WMMA 16x16x4.
// Block = 256 threads = 8 waves. Block b owns rows [16b, 16b+16).
// Wave w owns the 16-wide N-tile starting at 16w (8 waves cover N=128).
// A-tile (16x128 f32) staged in LDS with row stride 132 floats so that the
// strided per-lane fragment reads are bank-conflict free (132 mod 64 == 4).
// ---------------------------------------------------------------------------
#define LDS_STRIDE 132

__global__ __launch_bounds__(256)
void gcn_gemm_wmma_kernel(const float* __restrict__ support,
                          const float* __restrict__ weight,
                          const float* __restrict__ bias,
                          float*       __restrict__ out) {
    __shared__ float As[16 * LDS_STRIDE];

    const int mBase = blockIdx.x * 16;

    // Cooperative load of the 16x128 A tile (512 float4 by 256 threads).
    for (int i = threadIdx.x; i < 512; i += 256) {
        int r  = i >> 5;     // row 0..15
        int c4 = i & 31;     // float4 column 0..31
        float4 v = reinterpret_cast<const float4*>(
                       support + (size_t)(mBase + r) * D_IN)[c4];
        *reinterpret_cast<float4*>(&As[r * LDS_STRIDE + c4 * 4]) = v;
    }
    __syncthreads();

    const int wid   = threadIdx.x >> 5;   // wave id 0..7 -> N tile
    const int lane  = threadIdx.x & 31;
    const int half  = lane >> 4;          // 0: lanes 0-15, 1: lanes 16-31
    const int l16   = lane & 15;
    const int nBase = wid * 16;

    v8f acc = {};

    // K loop: 128 = 32 x (16x16x4 f32 WMMA).
    // A frag layout: lanes 0-15 -> M=lane,   v0=K0,   v1=K0+1
    //                lanes16-31 -> M=lane-16,v0=K0+2, v1=K0+3
    // B frag mirrors (rows striped across lanes within a VGPR).
#pragma unroll 8
    for (int k0 = 0; k0 < D_IN; k0 += 4) {
        const int ka = k0 + half * 2;

        v2f a;
        a.x = As[l16 * LDS_STRIDE + ka];
        a.y = As[l16 * LDS_STRIDE + ka + 1];

        const int n = nBase + l16;
        v2f b;
        b.x = weight[(size_t)ka * D_OUT + n];
        b.y = weight[(size_t)(ka + 1) * D_OUT + n];

        acc = __builtin_amdgcn_wmma_f32_16x16x4_f32(
            /*neg_a=*/false, a, /*neg_b=*/false, b,
            /*c_mod=*/(short)0, acc, /*reuse_a=*/false, /*reuse_b=*/false);
    }

    // C/D layout: VGPR r -> row (mBase + r + half*8), col (nBase + l16).
    const int   col = nBase + l16;
    const float bv  = bias[col];
#pragma unroll
    for (int r = 0; r < 8; ++r) {
        int row = mBase + r + half * 8;
        out[(size_t)row * D_OUT + col] = acc[r] + bv;
    }
}

// ---------------------------------------------------------------------------
extern "C" void kernel_launch(void* const* d_in, const int* in_sizes, int n_in,
                              void* d_out, int out_size, void* d_ws, size_t ws_size,
                              hipStream_t stream) {
    const float* x      = (const float*)d_in[0];
    const int*   src    = (const int*)  d_in[1];
    const int*   dst    = (const int*)  d_in[2];
    const float* w      = (const float*)d_in[3];
    const float* weight = (const float*)d_in[4];
    const float* bias   = (const float*)d_in[5];
    float*       out    = (float*)d_out;

    const int n_edges = in_sizes[1];

    float* support = (float*)d_ws;   // N_NODES * D_IN floats = 5.12 MB

    // Zero the accumulation buffer (graph-capture-legal memset node).
    hipMemsetAsync(support, 0, (size_t)N_NODES * D_IN * sizeof(float), stream);

    // Scatter: one wave32 per edge -> 8 edges per 256-thread block.
    int scatter_blocks = (n_edges + 7) / 8;
    gcn_scatter_kernel<<<scatter_blocks, 256, 0, stream>>>(
        x, src, dst, w, support, n_edges);

    // GEMM: 625 blocks x 16 rows, 8 waves x 16 cols.
    gcn_gemm_wmma_kernel<<<N_NODES / 16, 256, 0, stream>>>(
        support, weight, bias, out);
}